// SphericalHarmonics_56805237457290
// MI455X (gfx1250) — compile-verified
//
#include <hip/hip_runtime.h>
#include <hip/hip_bf16.h>

// Spherical harmonics l=0..3 (16 components) for N float3 vectors.
// Memory-bound streaming kernel: 4 edges/thread so that ALL global traffic is
// aligned B128, with non-temporal (TH_NT) hints since neither input nor the
// 256MB output has any reuse (output exceeds the 192MB L2).

typedef __attribute__((ext_vector_type(4))) float v4f;

// Integral-normalized real SH constants (match the JAX reference exactly).
#define C00  0.28209479177387814f
#define C1   0.4886025119029199f
#define C2M2 1.0925484305920792f
#define C20  0.31539156525252005f
#define C22  0.5462742152960396f
#define C3M3 0.5900435899266435f
#define C3M2 2.890611442640554f
#define C3M1 0.4570457994644658f
#define C30  0.3731763325901154f
#define C32  1.445305721320277f

__device__ __forceinline__ void sh16(float vx, float vy, float vz, float* o) {
    float n2 = vx * vx + vy * vy + vz * vz;
    float r  = __builtin_amdgcn_rsqf(n2);          // v_rsq_f32 (TRANS)
    float x = vx * r, y = vy * r, z = vz * r;
    float x2 = x * x, y2 = y * y, z2 = z * z;

    o[0]  = C00;                                   // l=0
    o[1]  = C1 * y;                                // l=1 m=-1
    o[2]  = C1 * z;                                // l=1 m=0
    o[3]  = C1 * x;                                // l=1 m=1
    o[4]  = C2M2 * x * y;                          // l=2 m=-2
    o[5]  = C2M2 * y * z;                          // l=2 m=-1
    o[6]  = C20 * (3.0f * z2 - 1.0f);              // l=2 m=0
    o[7]  = C2M2 * x * z;                          // l=2 m=1
    o[8]  = C22 * (x2 - y2);                       // l=2 m=2
    o[9]  = C3M3 * y * (3.0f * x2 - y2);           // l=3 m=-3
    o[10] = C3M2 * x * y * z;                      // l=3 m=-2
    o[11] = C3M1 * y * (5.0f * z2 - 1.0f);         // l=3 m=-1
    o[12] = C30 * z * (5.0f * z2 - 3.0f);          // l=3 m=0
    o[13] = C3M1 * x * (5.0f * z2 - 1.0f);         // l=3 m=1
    o[14] = C32 * z * (x2 - y2);                   // l=3 m=2
    o[15] = C3M3 * x * (x2 - 3.0f * y2);           // l=3 m=3
}

__global__ __launch_bounds__(256)
void sh_l3_kernel(const float* __restrict__ vecs,
                  float* __restrict__ out,
                  int n_edges) {
    const int t    = blockIdx.x * blockDim.x + threadIdx.x;
    const int base = t * 4;                        // first edge handled by this thread
    if (base >= n_edges) return;

    if (base + 4 <= n_edges) {
        // Fast path: 4 edges -> 3 aligned B128 NT loads, 16 B128 NT stores.
        const v4f* vp = (const v4f*)vecs;
        v4f a = __builtin_nontemporal_load(vp + 3 * t + 0);   // x0 y0 z0 x1
        v4f b = __builtin_nontemporal_load(vp + 3 * t + 1);   // y1 z1 x2 y2
        v4f c = __builtin_nontemporal_load(vp + 3 * t + 2);   // z2 x3 y3 z3

        float xs[4] = { a.x, a.w, b.z, c.y };
        float ys[4] = { a.y, b.x, b.w, c.z };
        float zs[4] = { a.z, b.y, c.x, c.w };

        #pragma unroll
        for (int e = 0; e < 4; ++e) {
            float o[16];
            sh16(xs[e], ys[e], zs[e], o);
            v4f* op = (v4f*)(out + (size_t)(base + e) * 16);
            v4f o0 = { o[0],  o[1],  o[2],  o[3]  };
            v4f o1 = { o[4],  o[5],  o[6],  o[7]  };
            v4f o2 = { o[8],  o[9],  o[10], o[11] };
            v4f o3 = { o[12], o[13], o[14], o[15] };
            __builtin_nontemporal_store(o0, op + 0);
            __builtin_nontemporal_store(o1, op + 1);
            __builtin_nontemporal_store(o2, op + 2);
            __builtin_nontemporal_store(o3, op + 3);
        }
    } else {
        // Tail (n_edges not a multiple of 4): scalar per-edge path.
        for (int e = base; e < n_edges; ++e) {
            float vx = vecs[3 * (size_t)e + 0];
            float vy = vecs[3 * (size_t)e + 1];
            float vz = vecs[3 * (size_t)e + 2];
            float o[16];
            sh16(vx, vy, vz, o);
            float* op = out + (size_t)e * 16;
            #pragma unroll
            for (int k = 0; k < 16; ++k) op[k] = o[k];
        }
    }
}

extern "C" void kernel_launch(void* const* d_in, const int* in_sizes, int n_in,
                              void* d_out, int out_size, void* d_ws, size_t ws_size,
                              hipStream_t stream) {
    const float* vecs = (const float*)d_in[0];
    float*       out  = (float*)d_out;

    const int n_edges = in_sizes[0] / 3;          // in_sizes[0] = 3 * N_EDGES
    const int threads_needed = (n_edges + 3) / 4; // 4 edges per thread
    const int block = 256;                        // 8 wave32 per block
    const int grid  = (threads_needed + block - 1) / block;

    sh_l3_kernel<<<grid, block, 0, stream>>>(vecs, out, n_edges);
}